// TensorProduct_1743756722477
// MI455X (gfx1250) — compile-verified
//
#include <hip/hip_runtime.h>

// e3nn-style tensor product, B=100000, MUL=128. Memory-bound streaming kernel:
// ~667 MB total traffic -> ~29 us floor at 23.3 TB/s HBM.
// Layout per batch element b:
//   x[b]  : 512 f32  = x0[128] ++ x1[128][3]
//   y[b]  : 4 f32    = y0 ++ y1[3]
//   W[b]  : 640 f32  = W0..W4 each [128]
//   out[b]: 512 f32  = out0[128] ++ out1[128][3]
// Strategy: one wave32 per element, each lane owns 4 consecutive u values so
// every global access is an aligned 128-bit NT load/store (b128, th:NT).

#define MUL 128
#define NPATH 5

typedef float v4f __attribute__((ext_vector_type(4)));

__global__ __launch_bounds__(256) void tp_e3nn_v4(
    const float* __restrict__ x,
    const float* __restrict__ y,
    const float* __restrict__ W,
    float* __restrict__ out,
    int Bn)
{
    const int tid  = threadIdx.x;
    const int b    = blockIdx.x * 8 + (tid >> 5);   // one wave32 per batch element
    if (b >= Bn) return;
    const int lane = tid & 31;                      // lane owns u = 4*lane .. 4*lane+3

    const float* xb = x   + (size_t)b * (4 * MUL);
    const float* Wb = W   + (size_t)b * (NPATH * MUL);
    float*       ob = out + (size_t)b * (4 * MUL);

    // y: 16 B, wave-uniform; keep default (cached) policy.
    const v4f yv = *(const v4f*)(y + (size_t)b * 4);
    const float y0  = yv.x;
    const float y1x = yv.y, y1y = yv.z, y1z = yv.w;

    // All streamed data as aligned b128 non-temporal loads.
    const v4f X0 = __builtin_nontemporal_load((const v4f*)xb + lane);

    const v4f* x1q = (const v4f*)(xb + MUL) + 3 * lane;   // 12 floats = 3 x b128
    const v4f XA = __builtin_nontemporal_load(x1q + 0);
    const v4f XB = __builtin_nontemporal_load(x1q + 1);
    const v4f XC = __builtin_nontemporal_load(x1q + 2);

    const v4f W0 = __builtin_nontemporal_load((const v4f*)(Wb + 0 * MUL) + lane);
    const v4f W1 = __builtin_nontemporal_load((const v4f*)(Wb + 1 * MUL) + lane);
    const v4f W2 = __builtin_nontemporal_load((const v4f*)(Wb + 2 * MUL) + lane);
    const v4f W3 = __builtin_nontemporal_load((const v4f*)(Wb + 3 * MUL) + lane);
    const v4f W4 = __builtin_nontemporal_load((const v4f*)(Wb + 4 * MUL) + lane);

    const float A0 = 0.70710678118654752f;   // sqrt(1/2)
    const float A1 = 0.40824829046386302f;   // sqrt(1/2)/sqrt(3)
    const float C3 = 0.57735026918962576f;   // 1/sqrt(3)
    const float C6 = 0.40824829046386302f;   // 1/sqrt(6)

    // Unpack the 4 sub-problems (all register-resident, fully unrolled).
    const float x0a[4]    = { X0.x, X0.y, X0.z, X0.w };
    const float x1a[4][3] = { { XA.x, XA.y, XA.z },
                              { XA.w, XB.x, XB.y },
                              { XB.z, XB.w, XC.x },
                              { XC.y, XC.z, XC.w } };
    const float w0a[4] = { W0.x, W0.y, W0.z, W0.w };
    const float w1a[4] = { W1.x, W1.y, W1.z, W1.w };
    const float w2a[4] = { W2.x, W2.y, W2.z, W2.w };
    const float w3a[4] = { W3.x, W3.y, W3.z, W3.w };
    const float w4a[4] = { W4.x, W4.y, W4.z, W4.w };

    const float ky0a = A0 * y0;   // hoisted per-element scalars
    const float kyb  = C3 * y0;

    v4f  O0;
    float o1[12];

#pragma unroll
    for (int j = 0; j < 4; ++j) {
        const float x0  = x0a[j];
        const float x1x = x1a[j][0], x1y = x1a[j][1], x1z = x1a[j][2];

        // out0
        const float dot = x1x * y1x + x1y * y1y + x1z * y1z;
        O0[j] = ky0a * (w0a[j] * x0) + A1 * (w1a[j] * dot);

        // cross(x1, y1)
        const float cx = x1y * y1z - x1z * y1y;
        const float cy = x1z * y1x - x1x * y1z;
        const float cz = x1x * y1y - x1y * y1x;

        const float ka = C3 * (w2a[j] * x0);   // * y1[i]
        const float kb = kyb * w3a[j];         // * x1[i]
        const float kc = C6 * w4a[j];          // * cross[i]

        o1[3 * j + 0] = ka * y1x + kb * x1x + kc * cx;
        o1[3 * j + 1] = ka * y1y + kb * x1y + kc * cy;
        o1[3 * j + 2] = ka * y1z + kb * x1z + kc * cz;
    }

    // Streaming stores: 4 aligned b128 NT stores per thread.
    __builtin_nontemporal_store(O0, (v4f*)ob + lane);

    v4f* oq = (v4f*)(ob + MUL) + 3 * lane;
    const v4f P0 = { o1[0], o1[1], o1[2],  o1[3]  };
    const v4f P1 = { o1[4], o1[5], o1[6],  o1[7]  };
    const v4f P2 = { o1[8], o1[9], o1[10], o1[11] };
    __builtin_nontemporal_store(P0, oq + 0);
    __builtin_nontemporal_store(P1, oq + 1);
    __builtin_nontemporal_store(P2, oq + 2);
}

extern "C" void kernel_launch(void* const* d_in, const int* in_sizes, int n_in,
                              void* d_out, int out_size, void* d_ws, size_t ws_size,
                              hipStream_t stream)
{
    const float* x = (const float*)d_in[0];
    const float* y = (const float*)d_in[1];
    const float* W = (const float*)d_in[2];
    float* out     = (float*)d_out;

    const int Bn = in_sizes[0] / (4 * MUL);        // 100000
    const int blocks = (Bn + 7) / 8;               // 8 elements (waves) per block

    tp_e3nn_v4<<<blocks, 256, 0, stream>>>(x, y, W, out, Bn);
}